// Model_11510512353959
// MI455X (gfx1250) — compile-verified
//
#include <hip/hip_runtime.h>
#include <hip/hip_bf16.h>
#include <math.h>

#define B_  128
#define L_  200
#define Lp_ 208          // L padded to a multiple of 16; pad rows kept at 0.0
#define D_  256
#define NT_ 10000

typedef __attribute__((ext_vector_type(2))) float v2f;
typedef __attribute__((ext_vector_type(8))) float v8f;

// ---------------------------------------------------------------------------
// Wave-tile fp32 GEMM on V_WMMA_F32_16X16X4_F32 (CDNA5 wave32).
// One wave owns a 16x16 tile of D = A(MxK)*B(KxN); K stepped by 4, fully
// unrolled (K, all leading dims, tile counts are template constants so every
// load is an immediate offset from one base pointer -> no per-iter address
// math, no divergence, EXEC all-ones for every WMMA).
// Frag layout per ISA 7.12.2:
//   A 16x4 : lane m = lane&15, K-half = lane>>4, 2 VGPRs (v2f, contiguous K)
//   B 4x16 : row K striped across lanes, halves split at lane 16
//   C/D    : VGPR v -> row v (+8 for lanes 16-31), col = lane&15
// BTRANS=1 reads B[k][n] = Bm[n*LDB + k] (contiguous in k -> b64 loads).
// ---------------------------------------------------------------------------
template<int Kc, int LDA, int LDB, int LDD, int NTILE, int TPB,
         int BTRANS, int LRELU>
__global__ __launch_bounds__(256)
void wmma_gemm_t(const float* __restrict__ A,
                 const float* __restrict__ Bm,
                 float* __restrict__ Dm,
                 long strideA, long strideB, long strideD)
{
    int wid   = (int)((blockIdx.x * blockDim.x + threadIdx.x) >> 5);
    int batch = wid / TPB;
    int t     = wid - batch * TPB;
    int rowT  = t / NTILE;
    int row0  = rowT * 16;
    int col0  = (t - rowT * NTILE) * 16;

    int lane = threadIdx.x & 31;
    int n16  = lane & 15;      // A row sub-index / B,D column sub-index
    int h    = lane >> 4;      // K-half for A/B, M-half for D

    const float* Arow = A + (long)batch * strideA
                          + (long)(row0 + n16) * LDA + 2 * h;
    const float* Bp;
    if (BTRANS)
        Bp = Bm + (long)batch * strideB + (long)(col0 + n16) * LDB + 2 * h;
    else
        Bp = Bm + (long)batch * strideB + (long)(2 * h) * LDB + col0 + n16;

    v8f acc = {0.f, 0.f, 0.f, 0.f, 0.f, 0.f, 0.f, 0.f};

#pragma unroll
    for (int k = 0; k < Kc; k += 4) {
        v2f a = *(const v2f*)(Arow + k);          // 8B aligned (even strides)
        v2f b;
        if (BTRANS) {
            b = *(const v2f*)(Bp + k);
        } else {
            b.x = Bp[(long)k * LDB];
            b.y = Bp[(long)k * LDB + LDB];
        }
        acc = __builtin_amdgcn_wmma_f32_16x16x4_f32(
                  false, a, false, b, (short)0, acc, false, false);
    }

    float* Drow = Dm + (long)batch * strideD + (long)row0 * LDD + col0 + n16;
#pragma unroll
    for (int v = 0; v < 8; ++v) {
        float x = acc[v];
        if (LRELU) x = x > 0.f ? x : 0.01f * x;
        Drow[(long)(v + 8 * h) * LDD] = x;
    }
}

// unorm[b][:] = user_emb[user_id[b]] / max(||.||2, 1e-12)
__global__ void user_norm_kernel(const int* __restrict__ user_id,
                                 const float* __restrict__ user_emb,
                                 float* __restrict__ unorm)
{
    __shared__ float red[256];
    int b = blockIdx.x, d = threadIdx.x;
    int uid = user_id[b];
    float v = user_emb[(long)uid * D_ + d];
    red[d] = v * v;
    __syncthreads();
    for (int s = 128; s > 0; s >>= 1) {
        if (d < s) red[d] += red[d + s];
        __syncthreads();
    }
    float nrm = sqrtf(red[0]);
    unorm[(long)b * D_ + d] = v / fmaxf(nrm, 1e-12f);
}

// enc (padded B x Lp x D): rows < L get E[et] + sign(E[et])*unorm[b],
// rows >= L get 0 (stay 0 through all layers). pop written for rows < L only.
__global__ void enc_pop_kernel(const int* __restrict__ event_type,
                               const float* __restrict__ E,
                               const float* __restrict__ P,
                               const float* __restrict__ unorm,
                               float* __restrict__ enc,
                               float* __restrict__ pop_out)
{
    int blp = blockIdx.x;            // b * Lp + l
    int b   = blp / Lp_;
    int l   = blp - b * Lp_;
    int d   = threadIdx.x;
    if (l >= L_) {                   // zero pad row
        enc[(long)blp * D_ + d] = 0.f;
        return;
    }
    int et = event_type[b * L_ + l];
    float np = (et != 0) ? 1.f : 0.f;
    float e  = E[(long)et * D_ + d];
    float u  = unorm[(long)b * D_ + d];
    float sg = (e > 0.f) ? 1.f : ((e < 0.f) ? -1.f : 0.f);
    enc[(long)blp * D_ + d] = e + sg * u;
    pop_out[((long)b * L_ + l) * D_ + d] = P[(long)et * D_ + d] * np;
}

// adj (padded B x Lp x L): row i<L: A[ri*NT+rj] + A[rj*NT+rj]; row i>=L: 0
__global__ void adj_kernel(const int* __restrict__ event_type,
                           const float* __restrict__ Adj,
                           float* __restrict__ out)
{
    long gid = (long)blockIdx.x * blockDim.x + threadIdx.x;  // over B*Lp*L
    int b   = (int)(gid / ((long)Lp_ * L_));
    int rem = (int)(gid - (long)b * Lp_ * L_);
    int i   = rem / L_;
    int j   = rem - i * L_;
    float v = 0.f;
    if (i < L_) {
        int ri = event_type[b * L_ + i] - 1;
        int rj = event_type[b * L_ + j] - 1;
        v = Adj[(long)ri * NT_ + rj] + Adj[(long)rj * NT_ + rj];
    }
    out[gid] = v;
}

// ue_out[b] = concat(mean_l enc, 0.1 * mean_l pop); ue_ws = first half
__global__ void mean_kernel(const float* __restrict__ enc,
                            const float* __restrict__ pop,
                            float* __restrict__ ue_out,
                            float* __restrict__ ue_ws)
{
    int b = blockIdx.x, d = threadIdx.x;
    const float* eb = enc + (long)b * Lp_ * D_ + d;
    const float* pb = pop + (long)b * L_ * D_ + d;
    float s = 0.f, p = 0.f;
    for (int l = 0; l < L_; ++l) {
        s += eb[(long)l * D_];
        p += pb[(long)l * D_];
    }
    s *= (1.f / (float)L_);
    p *= (1.f / (float)L_) * 0.1f;
    ue_out[(long)b * 2 * D_ + d]      = s;
    ue_out[(long)b * 2 * D_ + D_ + d] = p;
    ue_ws[(long)b * D_ + d]           = s;
}

// row-wise: pred = tanh(pred / max(||row||2, 1e-5))
__global__ void norm_tanh_kernel(float* __restrict__ pred)
{
    __shared__ float red[256];
    int b = blockIdx.x, tid = threadIdx.x;
    float* row = pred + (long)b * NT_;
    float s = 0.f;
    for (int t = tid; t < NT_; t += 256) {
        float v = row[t];
        s += v * v;
    }
    red[tid] = s;
    __syncthreads();
    for (int k = 128; k > 0; k >>= 1) {
        if (tid < k) red[tid] += red[tid + k];
        __syncthreads();
    }
    float inv = 1.f / fmaxf(sqrtf(red[0]), 1e-5f);
    for (int t = tid; t < NT_; t += 256) row[t] = tanhf(row[t] * inv);
}

extern "C" void kernel_launch(void* const* d_in, const int* in_sizes, int n_in,
                              void* d_out, int out_size, void* d_ws, size_t ws_size,
                              hipStream_t stream)
{
    const int*   user_id    = (const int*)d_in[0];
    const int*   event_type = (const int*)d_in[1];
    const float* adjm       = (const float*)d_in[2];
    const float* pop_enc    = (const float*)d_in[3];
    const float* event_emb  = (const float*)d_in[4];
    const float* user_emb   = (const float*)d_in[5];
    const float* layer_w    = (const float*)d_in[6];

    float* out     = (float*)d_out;
    float* pred    = out;                              // B * NT
    float* ue_out  = out + (long)B_ * NT_;             // B * 2D
    float* pop_out = ue_out + (long)B_ * 2 * D_;       // B * L * D

    float* ws    = (float*)d_ws;
    float* unorm = ws;                                 // B*D
    float* ue_ws = unorm + (long)B_ * D_;              // B*D
    float* enc   = ue_ws + (long)B_ * D_;              // B*Lp*D
    float* tmp   = enc   + (long)B_ * Lp_ * D_;        // B*Lp*D
    float* adj   = tmp   + (long)B_ * Lp_ * D_;        // B*Lp*L  (~76 MB total)

    user_norm_kernel<<<B_, 256, 0, stream>>>(user_id, user_emb, unorm);
    enc_pop_kernel<<<B_ * Lp_, 256, 0, stream>>>(event_type, event_emb, pop_enc,
                                                 unorm, enc, pop_out);
    adj_kernel<<<(int)(((long)B_ * Lp_ * L_) / 256), 256, 0, stream>>>(
        event_type, adjm, adj);

    // 13 x 16 tiles per batch, 8 waves/block -> exact grids, no guards
    const int TPB1   = (Lp_ / 16) * (D_ / 16);         // 13*16 = 208
    const int tiles1 = B_ * TPB1;                      // 26624
    const int blk1   = tiles1 / 8;                     // 3328

    for (int i = 0; i < 4; ++i) {
        // tmp = adj @ enc   (M=208 padded, N=256, K=200)
        wmma_gemm_t<L_, L_, D_, D_, D_ / 16, TPB1, 0, 0>
            <<<blk1, 256, 0, stream>>>(
                adj, enc, tmp,
                (long)Lp_ * L_, (long)Lp_ * D_, (long)Lp_ * D_);
        // enc = leaky_relu(tmp @ W_i)   (M=208 padded, N=256, K=256)
        wmma_gemm_t<D_, D_, D_, D_, D_ / 16, TPB1, 0, 1>
            <<<blk1, 256, 0, stream>>>(
                tmp, layer_w + (long)i * D_ * D_, enc,
                (long)Lp_ * D_, 0L, (long)Lp_ * D_);
    }

    mean_kernel<<<B_, 256, 0, stream>>>(enc, pop_out, ue_out, ue_ws);

    // pred = ue_ws @ event_emb[1:].T   (M=128, N=10000, K=256), B transposed
    const int TPB2   = (B_ / 16) * (NT_ / 16);         // 8*625 = 5000
    wmma_gemm_t<D_, D_, D_, NT_, NT_ / 16, TPB2, 1, 0>
        <<<TPB2 / 8, 256, 0, stream>>>(
            ue_ws, event_emb + D_, pred, 0L, 0L, 0L);

    norm_tanh_kernel<<<B_, 256, 0, stream>>>(pred);
}